// StrucFeaGNN_68410239091211
// MI455X (gfx1250) — compile-verified
//
#include <hip/hip_runtime.h>
#include <hip/hip_bf16.h>

typedef __attribute__((ext_vector_type(2))) float v2f;
typedef __attribute__((ext_vector_type(8))) float v8f;

#define N_NODES 100000
#define N_EDGES 800000
#define N_GRAPHS 64
#define HDIM 32      // H = 32
#define INDIM 144    // IDENT + 2*ED

// ---------------------------------------------------------------------------
// Generic fp32 zero-fill (grid-stride).
// ---------------------------------------------------------------------------
__global__ __launch_bounds__(256) void zero_f32(float* __restrict__ p, long n) {
  long i = (long)blockIdx.x * blockDim.x + threadIdx.x;
  long stride = (long)gridDim.x * blockDim.x;
  for (; i < n; i += stride) p[i] = 0.0f;
}

// ---------------------------------------------------------------------------
// Preprocess: new_x[i] = concat(ident(128), relu-MLP-on-struct(16)).
// One wave per node; lanes 0..15 compute the 16 embedding columns.
// ---------------------------------------------------------------------------
__global__ __launch_bounds__(256) void preproc_kernel(
    const float* __restrict__ x,
    const float* __restrict__ pw0, const float* __restrict__ pb0,
    const float* __restrict__ pw1, const float* __restrict__ pb1,
    const float* __restrict__ ew,  const float* __restrict__ eb,
    float* __restrict__ newx) {
  int wave = threadIdx.x >> 5;
  int lane = threadIdx.x & 31;
  int node = blockIdx.x * 8 + wave;
  if (node >= N_NODES) return;

  const float* xr = x + (long)node * 130;
  float* o = newx + (long)node * INDIM;

  // copy ident (128 floats) cooperatively
  for (int c = lane; c < 128; c += 32) o[c] = xr[c];

  if (lane < 16) {
    float s0 = xr[128];
    float s1 = xr[129];
    float acc = eb[lane];
#pragma unroll
    for (int k = 0; k < 8; ++k) {
      float h0 = fmaxf(s0 * pw0[k] + pb0[k], 0.0f);
      acc += h0 * ew[k * 16 + lane];
    }
#pragma unroll
    for (int k = 0; k < 8; ++k) {
      float h1 = fmaxf(s1 * pw1[k] + pb1[k], 0.0f);
      acc += h1 * ew[(8 + k) * 16 + lane];
    }
    o[128 + lane] = acc;
  }
}

// ---------------------------------------------------------------------------
// Edge scatter-add: agg[dst] += feat[src], float4 chunks + global f32 atomics.
// Bandwidth/atomic bound; agg fits in the 192MB L2 so atomics resolve there.
// ---------------------------------------------------------------------------
template <int F>
__global__ __launch_bounds__(256) void scatter_add_kernel(
    const int* __restrict__ src, const int* __restrict__ dst,
    const float* __restrict__ feat, float* __restrict__ agg) {
  const int CH = F / 4;
  long t = (long)blockIdx.x * blockDim.x + threadIdx.x;
  long e = t / CH;
  int c = (int)(t % CH);
  if (e >= N_EDGES) return;
  int s = src[e];
  int d = dst[e];
  const float4 v = *(const float4*)(feat + (long)s * F + c * 4);
  float* out = agg + (long)d * F + c * 4;
  atomicAdd(out + 0, v.x);
  atomicAdd(out + 1, v.y);
  atomicAdd(out + 2, v.z);
  atomicAdd(out + 3, v.w);
}

// ---------------------------------------------------------------------------
// Fused GIN layer:
//   Z = (h + agg) @ W1 + b1      (WMMA f32 16x16x4, K = 144 or 32)
//   Z = BN(Z); R = relu(Z)       (register epilogue)
//   Y = R @ W2 + b2              (WMMA via per-wave LDS re-layout)
//   out = mode0: relu(Y)
//         mode1: relu(Y) + res0
//         mode2: Y + res0 + res1
// One wave owns a 16-row x 32-col tile: two v8f accumulators.
// ---------------------------------------------------------------------------
__global__ __launch_bounds__(256) void gin_layer_kernel(
    const float* __restrict__ hIn, const float* __restrict__ agg, int K,
    const float* __restrict__ w1, const float* __restrict__ b1,
    const float* __restrict__ bng, const float* __restrict__ bnb,
    const float* __restrict__ rm, const float* __restrict__ rv,
    const float* __restrict__ w2, const float* __restrict__ b2,
    const float* __restrict__ res0, const float* __restrict__ res1,
    float* __restrict__ out, int mode) {
  __shared__ float lds[8][16][33];   // per-wave 16x32 R tile, +1 pad vs bank conflicts

  int wave = threadIdx.x >> 5;
  int lane = threadIdx.x & 31;
  int wt = blockIdx.x * 8 + wave;
  int m0 = wt * 16;
  if (m0 >= N_NODES) return;         // whole-wave exit; EXEC stays all-1s for WMMA

  int row  = lane & 15;              // A-fragment row / LDS row
  int half = lane >> 4;              // 0 or 1
  int kh   = half * 2;               // K sub-offset within a step of 4
  int n0   = lane & 15;              // C/D column for acc0
  int n1   = n0 + 16;                // C/D column for acc1

  // ---- GEMM 1: Z = (h + agg) @ W1 -------------------------------------
  v8f acc0 = {};
  v8f acc1 = {};
  const float* hr = hIn + (long)(m0 + row) * K;
  const float* ar = agg + (long)(m0 + row) * K;
  for (int k0 = 0; k0 < K; k0 += 4) {
    v2f a;
    a.x = hr[k0 + kh]     + ar[k0 + kh];
    a.y = hr[k0 + kh + 1] + ar[k0 + kh + 1];
    v2f bf0, bf1;
    bf0.x = w1[(k0 + kh) * HDIM + n0];
    bf0.y = w1[(k0 + kh + 1) * HDIM + n0];
    bf1.x = w1[(k0 + kh) * HDIM + n1];
    bf1.y = w1[(k0 + kh + 1) * HDIM + n1];
    acc0 = __builtin_amdgcn_wmma_f32_16x16x4_f32(false, a, false, bf0, (short)0, acc0, false, false);
    acc1 = __builtin_amdgcn_wmma_f32_16x16x4_f32(false, a, false, bf1, (short)0, acc1, false, false);
  }

  // ---- BN + relu epilogue in registers, spill R tile to LDS -----------
  float sc0  = rsqrtf(rv[n0] + 1e-5f) * bng[n0];
  float sc1  = rsqrtf(rv[n1] + 1e-5f) * bng[n1];
  float off0 = (b1[n0] - rm[n0]) * sc0 + bnb[n0];
  float off1 = (b1[n1] - rm[n1]) * sc1 + bnb[n1];
#pragma unroll
  for (int r = 0; r < 8; ++r) {
    int mr = r + 8 * half;
    lds[wave][mr][n0] = fmaxf(acc0[r] * sc0 + off0, 0.0f);
    lds[wave][mr][n1] = fmaxf(acc1[r] * sc1 + off1, 0.0f);
  }

  // ---- GEMM 2: Y = R @ W2 (K = 32) ------------------------------------
  v8f y0 = {};
  v8f y1 = {};
#pragma unroll
  for (int k0 = 0; k0 < HDIM; k0 += 4) {
    v2f a;
    a.x = lds[wave][row][k0 + kh];
    a.y = lds[wave][row][k0 + kh + 1];
    v2f bf0, bf1;
    bf0.x = w2[(k0 + kh) * HDIM + n0];
    bf0.y = w2[(k0 + kh + 1) * HDIM + n0];
    bf1.x = w2[(k0 + kh) * HDIM + n1];
    bf1.y = w2[(k0 + kh + 1) * HDIM + n1];
    y0 = __builtin_amdgcn_wmma_f32_16x16x4_f32(false, a, false, bf0, (short)0, y0, false, false);
    y1 = __builtin_amdgcn_wmma_f32_16x16x4_f32(false, a, false, bf1, (short)0, y1, false, false);
  }

  // ---- bias + activation + residual store -----------------------------
#pragma unroll
  for (int r = 0; r < 8; ++r) {
    int node = m0 + r + 8 * half;
    long base = (long)node * HDIM;
    float v0 = y0[r] + b2[n0];
    float v1 = y1[r] + b2[n1];
    if (mode == 0) {
      v0 = fmaxf(v0, 0.0f);
      v1 = fmaxf(v1, 0.0f);
    } else if (mode == 1) {
      v0 = fmaxf(v0, 0.0f) + res0[base + n0];
      v1 = fmaxf(v1, 0.0f) + res0[base + n1];
    } else {
      v0 += res0[base + n0] + res1[base + n0];
      v1 += res0[base + n1] + res1[base + n1];
    }
    out[base + n0] = v0;
    out[base + n1] = v1;
  }
}

// ---------------------------------------------------------------------------
// Graph pooling: per-graph sums + counts via atomics.
// ---------------------------------------------------------------------------
__global__ __launch_bounds__(256) void pool_kernel(
    const float* __restrict__ ge2, const int* __restrict__ batch,
    float* __restrict__ sums, float* __restrict__ cnt) {
  long t = (long)blockIdx.x * blockDim.x + threadIdx.x;
  if (t >= (long)N_NODES * HDIM) return;
  int node = (int)(t >> 5);
  int f = (int)(t & 31);
  int b = batch[node];
  atomicAdd(&sums[b * HDIM + f], ge2[t]);
  if (f == 0) atomicAdd(&cnt[b], 1.0f);
}

// ---------------------------------------------------------------------------
// Head: mean, 32->16 relu MLP, 16->7, log-softmax. One thread per graph.
// ---------------------------------------------------------------------------
__global__ __launch_bounds__(64) void head_kernel(
    const float* __restrict__ sums, const float* __restrict__ cnt,
    const float* __restrict__ p1w, const float* __restrict__ p1b,
    const float* __restrict__ p2w, const float* __restrict__ p2b,
    float* __restrict__ out) {
  int g = threadIdx.x;
  if (g >= N_GRAPHS) return;
  float c = fmaxf(cnt[g], 1.0f);
  float tmp[HDIM];
#pragma unroll
  for (int f = 0; f < HDIM; ++f) tmp[f] = sums[g * HDIM + f] / c;
  float hid[16];
#pragma unroll
  for (int j = 0; j < 16; ++j) {
    float acc = p1b[j];
    for (int f = 0; f < HDIM; ++f) acc += tmp[f] * p1w[f * 16 + j];
    hid[j] = fmaxf(acc, 0.0f);
  }
  float logits[7];
  float mx = -1e30f;
#pragma unroll
  for (int o = 0; o < 7; ++o) {
    float acc = p2b[o];
    for (int j = 0; j < 16; ++j) acc += hid[j] * p2w[j * 7 + o];
    logits[o] = acc;
    mx = fmaxf(mx, acc);
  }
  float se = 0.0f;
#pragma unroll
  for (int o = 0; o < 7; ++o) se += __expf(logits[o] - mx);
  float lse = __logf(se) + mx;
#pragma unroll
  for (int o = 0; o < 7; ++o) out[g * 7 + o] = logits[o] - lse;
}

// ---------------------------------------------------------------------------
extern "C" void kernel_launch(void* const* d_in, const int* in_sizes, int n_in,
                              void* d_out, int out_size, void* d_ws, size_t ws_size,
                              hipStream_t stream) {
  (void)in_sizes; (void)n_in; (void)out_size; (void)ws_size;

  const float* x      = (const float*)d_in[0];
  const int*   src    = (const int*)d_in[1];
  const int*   dst    = (const int*)d_in[2];
  const int*   batch  = (const int*)d_in[3];
  const float* pre_w0 = (const float*)d_in[4];
  const float* pre_b0 = (const float*)d_in[5];
  const float* pre_w1 = (const float*)d_in[6];
  const float* pre_b1 = (const float*)d_in[7];
  const float* e_w    = (const float*)d_in[8];
  const float* e_b    = (const float*)d_in[9];
  const float* p1w    = (const float*)d_in[10];
  const float* p1b    = (const float*)d_in[11];
  const float* p2w    = (const float*)d_in[12];
  const float* p2b    = (const float*)d_in[13];
  // per-layer params start at 14, 8 tensors each: w1,b1,g,b,rm,rv,w2,b2
  const float* gp[3][8];
  for (int l = 0; l < 3; ++l)
    for (int t = 0; t < 8; ++t) gp[l][t] = (const float*)d_in[14 + l * 8 + t];

  float* out = (float*)d_out;

  // workspace layout (floats)
  float* ws   = (float*)d_ws;
  float* newx = ws;                           // 100000*144
  float* agg  = newx + (long)N_NODES * INDIM; // 100000*144 (reused 32-wide)
  float* ge0  = agg + (long)N_NODES * INDIM;  // 100000*32
  float* ge1  = ge0 + (long)N_NODES * HDIM;
  float* ge2  = ge1 + (long)N_NODES * HDIM;
  float* sums = ge2 + (long)N_NODES * HDIM;   // 64*32
  float* cnt  = sums + N_GRAPHS * HDIM;       // 64

  const int ginBlocks = ((N_NODES / 16) + 7) / 8;  // 6250 wave tiles, 8 waves/block

  // 1. preprocess
  preproc_kernel<<<(N_NODES + 7) / 8, 256, 0, stream>>>(
      x, pre_w0, pre_b0, pre_w1, pre_b1, e_w, e_b, newx);

  // 2. layer 0 (K=144)
  zero_f32<<<4096, 256, 0, stream>>>(agg, (long)N_NODES * INDIM);
  {
    long tot = (long)N_EDGES * (INDIM / 4);
    scatter_add_kernel<INDIM><<<(unsigned)((tot + 255) / 256), 256, 0, stream>>>(
        src, dst, newx, agg);
  }
  gin_layer_kernel<<<ginBlocks, 256, 0, stream>>>(
      newx, agg, INDIM, gp[0][0], gp[0][1], gp[0][2], gp[0][3], gp[0][4], gp[0][5],
      gp[0][6], gp[0][7], nullptr, nullptr, ge0, 0);

  // 3. layer 1 (K=32)
  zero_f32<<<4096, 256, 0, stream>>>(agg, (long)N_NODES * HDIM);
  {
    long tot = (long)N_EDGES * (HDIM / 4);
    scatter_add_kernel<HDIM><<<(unsigned)((tot + 255) / 256), 256, 0, stream>>>(
        src, dst, ge0, agg);
  }
  gin_layer_kernel<<<ginBlocks, 256, 0, stream>>>(
      ge0, agg, HDIM, gp[1][0], gp[1][1], gp[1][2], gp[1][3], gp[1][4], gp[1][5],
      gp[1][6], gp[1][7], ge0, nullptr, ge1, 1);

  // 4. layer 2 (K=32)
  zero_f32<<<4096, 256, 0, stream>>>(agg, (long)N_NODES * HDIM);
  {
    long tot = (long)N_EDGES * (HDIM / 4);
    scatter_add_kernel<HDIM><<<(unsigned)((tot + 255) / 256), 256, 0, stream>>>(
        src, dst, ge1, agg);
  }
  gin_layer_kernel<<<ginBlocks, 256, 0, stream>>>(
      ge1, agg, HDIM, gp[2][0], gp[2][1], gp[2][2], gp[2][3], gp[2][4], gp[2][5],
      gp[2][6], gp[2][7], ge0, ge1, ge2, 2);

  // 5. pooling + head
  zero_f32<<<16, 256, 0, stream>>>(sums, N_GRAPHS * HDIM + N_GRAPHS);
  {
    long tot = (long)N_NODES * HDIM;
    pool_kernel<<<(unsigned)((tot + 255) / 256), 256, 0, stream>>>(ge2, batch, sums, cnt);
  }
  head_kernel<<<1, 64, 0, stream>>>(sums, cnt, p1w, p1b, p2w, p2b, out);
}